// YoloLoss_59691455480538
// MI455X (gfx1250) — compile-verified
//
#include <hip/hip_runtime.h>
#include <hip/hip_bf16.h>

// YOLO loss (forward) for MI455X / gfx1250.
// Bandwidth-bound streaming reduction: ~172 MB in, 5 floats out.
// Pass 1: 1024 blocks x 256 threads grid-stride over N*S*S cells, per-thread
//         accumulation of 4 partial sums, wave32 reduction via
//         V_WMMA_F32_16X16X4_F32 (ones-matrix row-sum trick), LDS across the
//         8 waves, per-block partials to d_ws.
// Pass 2: 1 block, 4 waves, each wave reduces one component over the 1024
//         block partials (WMMA wave reduction again), thread 0 composes the
//         5 outputs ([total, reg, contain, noobj, cls] / N).

typedef float f2  __attribute__((ext_vector_type(2)));
typedef float f4  __attribute__((ext_vector_type(4)));
typedef float v8f __attribute__((ext_vector_type(8)));

#define NBLOCKS  1024
#define NTHREADS 256
#define S_INV    (1.0f / 28.0f)
#define L_COORD  5.0f
#define L_NOOBJ  0.5f

// Full 32-lane sum using the matrix pipe.
// A-vgpr0 = per-lane value v, A-vgpr1 = 0, B = ones(4x16).
// D[m][n] = sum_k A[m][k] = v_m + v_{m+16}  (A 16x4 layout, ISA 7.12.2).
// D layout: vgpr r, lanes 0-15 -> D[r][lane]; lanes 16-31 -> D[8+r][lane-16].
// So sum(c[0..7]) gives lanes0-15: rows 0..7 = lanes {0..7,16..23};
// lanes16-31: rows 8..15 = lanes {8..15,24..31}. One xor-16 add completes it.
// Requires EXEC == all ones (all call sites are fully convergent).
__device__ __forceinline__ float wave_sum32(float v) {
    f2 a;    a.x = v;    a.y = 0.0f;
    f2 ones; ones.x = 1.0f; ones.y = 1.0f;
    v8f c = {0.0f, 0.0f, 0.0f, 0.0f, 0.0f, 0.0f, 0.0f, 0.0f};
    c = __builtin_amdgcn_wmma_f32_16x16x4_f32(
        /*neg_a=*/false, a, /*neg_b=*/false, ones,
        /*c_mod=*/(short)0, c, /*reuse_a=*/false, /*reuse_b=*/false);
    float s = ((c[0] + c[1]) + (c[2] + c[3])) + ((c[4] + c[5]) + (c[6] + c[7]));
    s += __shfl_xor(s, 16, 32);
    return s;  // full-wave sum, broadcast to all lanes
}

struct Box { float x1, y1, x2, y2; };

__device__ __forceinline__ Box to_xyxy(float b0, float b1, float w, float h) {
    float cx = b0 * S_INV, cy = b1 * S_INV;
    Box r;
    r.x1 = cx - 0.5f * w; r.y1 = cy - 0.5f * h;
    r.x2 = cx + 0.5f * w; r.y2 = cy + 0.5f * h;
    return r;
}

__device__ __forceinline__ float iou(const Box& a, const Box& b) {
    float lx = fmaxf(a.x1, b.x1), ly = fmaxf(a.y1, b.y1);
    float rx = fminf(a.x2, b.x2), ry = fminf(a.y2, b.y2);
    float w = fmaxf(rx - lx, 0.0f), h = fmaxf(ry - ly, 0.0f);
    float inter = w * h;
    float a1 = (a.x2 - a.x1) * (a.y2 - a.y1);
    float a2 = (b.x2 - b.x1) * (b.y2 - b.y1);
    float den = a1 + a2 - inter;
    return inter / (den > 0.0f ? den : 1.0f);
}

__global__ __launch_bounds__(NTHREADS)
void yolo_partial(const float* __restrict__ pred,
                  const float* __restrict__ tbox,
                  const float* __restrict__ tcls,
                  const unsigned char* __restrict__ hasobj,
                  float* __restrict__ partials,
                  int total) {
    float acc_cls = 0.0f, acc_noobj = 0.0f, acc_cont = 0.0f, acc_xywh = 0.0f;
    const int stride = gridDim.x * blockDim.x;
    for (int i = blockIdx.x * blockDim.x + threadIdx.x; i < total; i += stride) {
        const f2* p2 = (const f2*)pred + (size_t)i * 15;  // 30 floats/cell, 8B aligned
        // Confidence words first: p[4] in w2.x, p[9] in w4.y (needed either way).
        f2 w2 = __builtin_nontemporal_load(p2 + 2);   // p4, p5
        f2 w4 = __builtin_nontemporal_load(p2 + 4);   // p8, p9
        bool m = hasobj[i] != 0;
        if (m) {
            f2 w0 = __builtin_nontemporal_load(p2 + 0);   // p0, p1
            f2 w1 = __builtin_nontemporal_load(p2 + 1);   // p2, p3
            f2 w3 = __builtin_nontemporal_load(p2 + 3);   // p6, p7
            // ---- class loss: sum over 20 channels (p[10..29]) ----
            float pc[20];
            #pragma unroll
            for (int k = 0; k < 10; ++k) {
                f2 t = __builtin_nontemporal_load(p2 + 5 + k);
                pc[2 * k] = t.x; pc[2 * k + 1] = t.y;
            }
            const f4* c4 = (const f4*)tcls + (size_t)i * 5;  // 20 floats, 16B aligned
            float cls = 0.0f;
            #pragma unroll
            for (int k = 0; k < 5; ++k) {
                f4 t = __builtin_nontemporal_load(c4 + k);
                float d0 = pc[4 * k + 0] - t.x;
                float d1 = pc[4 * k + 1] - t.y;
                float d2 = pc[4 * k + 2] - t.z;
                float d3 = pc[4 * k + 3] - t.w;
                cls += d0 * d0 + d1 * d1 + d2 * d2 + d3 * d3;
            }
            acc_cls += cls;
            // ---- box terms ----
            f4 tb = __builtin_nontemporal_load((const f4*)tbox + i);
            float b1x = w0.x, b1y = w0.y, b1w = w1.x, b1h = w1.y, b1c = w2.x;
            float b2x = w2.y, b2y = w3.x, b2w = w3.y, b2h = w4.x, b2c = w4.y;
            Box tg = to_xyxy(tb.x, tb.y, tb.z, tb.w);
            float i1 = iou(to_xyxy(b1x, b1y, b1w, b1h), tg);
            float i2 = iou(to_xyxy(b2x, b2y, b2w, b2h), tg);
            bool take1 = i1 > i2;
            float best_iou = take1 ? i1 : i2;
            float bx = take1 ? b1x : b2x;
            float by = take1 ? b1y : b2y;
            float bw = take1 ? b1w : b2w;
            float bh = take1 ? b1h : b2h;
            float bc = take1 ? b1c : b2c;
            float dx = bx - tb.x, dy = by - tb.y;
            float sw = sqrtf(bw) - sqrtf(tb.z);
            float sh = sqrtf(bh) - sqrtf(tb.w);
            acc_xywh += dx * dx + dy * dy + sw * sw + sh * sh;
            float dc = bc - best_iou;
            acc_cont += dc * dc;
        } else {
            acc_noobj += w2.x * w2.x + w4.y * w4.y;  // p4^2 + p9^2
        }
    }

    // Fully convergent here (EXEC all ones) -> wave32 WMMA reductions.
    float s0 = wave_sum32(acc_cls);
    float s1 = wave_sum32(acc_noobj);
    float s2 = wave_sum32(acc_cont);
    float s3 = wave_sum32(acc_xywh);

    __shared__ float sm[NTHREADS / 32][4];
    int wave = threadIdx.x >> 5, lane = threadIdx.x & 31;
    if (lane == 0) { sm[wave][0] = s0; sm[wave][1] = s1; sm[wave][2] = s2; sm[wave][3] = s3; }
    __syncthreads();
    if (threadIdx.x < 4) {
        float t = 0.0f;
        #pragma unroll
        for (int w = 0; w < NTHREADS / 32; ++w) t += sm[w][threadIdx.x];
        partials[blockIdx.x * 4 + threadIdx.x] = t;
    }
}

__global__ __launch_bounds__(128)
void yolo_final(const float* __restrict__ partials, float* __restrict__ out, float invN) {
    int wave = threadIdx.x >> 5, lane = threadIdx.x & 31;  // wave = component 0..3
    float t = 0.0f;
    #pragma unroll 4
    for (int j = lane; j < NBLOCKS; j += 32) t += partials[j * 4 + wave];
    float s = wave_sum32(t);   // all 128 threads active, uniform trip count
    __shared__ float comp[4];
    if (lane == 0) comp[wave] = s;
    __syncthreads();
    if (threadIdx.x == 0) {
        float cls = comp[0];
        float noo = comp[1] * L_NOOBJ;
        float con = comp[2];
        float reg = comp[3] * L_COORD;
        float tot = cls + noo + con + reg;
        out[0] = tot * invN;
        out[1] = reg * invN;
        out[2] = con * invN;
        out[3] = noo * invN;
        out[4] = cls * invN;
    }
}

extern "C" void kernel_launch(void* const* d_in, const int* in_sizes, int n_in,
                              void* d_out, int out_size, void* d_ws, size_t ws_size,
                              hipStream_t stream) {
    const float* pred = (const float*)d_in[0];                // (N,28,28,30) f32
    const float* tbox = (const float*)d_in[1];                // (N,28,28,4)  f32
    const float* tcls = (const float*)d_in[2];                // (N,28,28,20) f32
    const unsigned char* mask = (const unsigned char*)d_in[3];// (N,28,28) bool (1 byte)
    int total = in_sizes[3];                                  // N*S*S cells
    float invN = 784.0f / (float)total;                       // 1/N
    float* partials = (float*)d_ws;                           // NBLOCKS*4 floats

    yolo_partial<<<NBLOCKS, NTHREADS, 0, stream>>>(pred, tbox, tcls, mask, partials, total);
    yolo_final<<<1, 128, 0, stream>>>(partials, (float*)d_out, invN);
}